// MultiBoxLoss_84335977824375
// MI455X (gfx1250) — compile-verified
//
#include <hip/hip_runtime.h>
#include <hip/hip_bf16.h>

// Problem constants (match reference)
#define B_   64
#define P_   8732
#define C_   81
#define O_   32
#define BP   (B_ * P_)
#define THRESH 0.5f
#define TILE 64

// Try CDNA5 async global->LDS staging (ASYNCcnt path). Set to 0 to fall back
// to vectorized VGPR round-trip staging.
#define USE_ASYNC_LDS 1

static_assert(BP % TILE == 0, "tile must divide rows");
static_assert((TILE * C_) % 4 == 0, "tile bytes must be 16B-multiple");
static_assert(P_ % 4 == 0, "row must be 16B-multiple");

typedef float v2f __attribute__((ext_vector_type(2)));
typedef float v8f __attribute__((ext_vector_type(8)));

#if __has_builtin(__builtin_amdgcn_wmma_f32_16x16x4_f32)
#define HAVE_WMMA_F32X4 1
#else
#define HAVE_WMMA_F32X4 0
#endif

// ---------------------------------------------------------------- init
__global__ void k_init(unsigned long long* __restrict__ best) {
    int i = blockIdx.x * blockDim.x + threadIdx.x;
    if (i < B_ * O_) best[i] = 0ull;
}

// ---------------------------------------------------------------- match
// Per (b,p): argmax IoU over 32 objects. Per (b,o): best prior via packed
// u64 atomicMax (iou_bits << 32 | (0xFFFFFFFF - p)) -> first-index tiebreak.
__global__ __launch_bounds__(256) void k_match(
    const float* __restrict__ tboxes,   // [B,O,4] boundary
    const float* __restrict__ pboxes,   // [P,4] center
    float* __restrict__ obj_ov,         // [B,P]
    int*   __restrict__ obj_idx,        // [B,P]
    unsigned long long* __restrict__ best) {  // [B,O]
    __shared__ float tb[O_ * 4];
    __shared__ unsigned long long lb[O_];
    const int b   = blockIdx.y;
    const int tid = threadIdx.x;
    if (tid < O_ * 4) tb[tid] = tboxes[b * O_ * 4 + tid];
    if (tid < O_)     lb[tid] = 0ull;
    __syncthreads();

    const int  p     = blockIdx.x * 256 + tid;
    const bool valid = (p < P_);
    float px0 = 0.f, py0 = 0.f, px1 = 0.f, py1 = 0.f, pw = 1.f, ph = 1.f;
    if (valid) {
        float4 q = ((const float4*)pboxes)[p];          // one b128 load
        pw  = q.z; ph = q.w;
        px0 = q.x - 0.5f * pw; py0 = q.y - 0.5f * ph;
        px1 = q.x + 0.5f * pw; py1 = q.y + 0.5f * ph;
    }
    const float areap = pw * ph;

    float bestov = -1.0f;
    int   besto  = 0;
    for (int o = 0; o < O_; ++o) {
        float ax0 = tb[o * 4 + 0], ay0 = tb[o * 4 + 1];
        float ax1 = tb[o * 4 + 2], ay1 = tb[o * 4 + 3];
        float iw    = fminf(ax1, px1) - fmaxf(ax0, px0);
        float ih    = fminf(ay1, py1) - fmaxf(ay0, py0);
        float inter = fmaxf(iw, 0.f) * fmaxf(ih, 0.f);
        float areaa = (ax1 - ax0) * (ay1 - ay0);
        float iou   = inter / (areaa + areap - inter);
        if (valid) {
            if (iou > bestov) { bestov = iou; besto = o; }  // strict > : first max
            unsigned long long pk =
                ((unsigned long long)__float_as_uint(iou) << 32) |
                (unsigned long long)(0xFFFFFFFFu - (unsigned)p);
            atomicMax(&lb[o], pk);
        }
    }
    __syncthreads();
    if (tid < O_) atomicMax(&best[b * O_ + tid], lb[tid]);
    if (valid) {
        obj_ov[b * P_ + p]  = bestov;
        obj_idx[b * P_ + p] = besto;
    }
}

// ---------------------------------------------------------------- forced match
__global__ void k_force(const unsigned long long* __restrict__ best,
                        float* __restrict__ obj_ov, int* __restrict__ obj_idx) {
    const int b = blockIdx.x;
    if (threadIdx.x != 0) return;
    for (int o = 0; o < O_; ++o) {           // serial: last write wins
        unsigned long long v = best[b * O_ + o];
        unsigned p = 0xFFFFFFFFu - (unsigned)(v & 0xFFFFFFFFull);
        if (p < P_) {
            obj_idx[b * P_ + p] = o;
            obj_ov[b * P_ + p]  = 1.0f;
        }
    }
}

// ---------------------------------------------------------------- labels + loc loss
__global__ __launch_bounds__(256) void k_labels(
    const float* __restrict__ pred_boxes,   // [B,P,4]
    const float* __restrict__ tboxes,       // [B,O,4]
    const int*   __restrict__ tcls,         // [B,O]
    const float* __restrict__ pboxes,       // [P,4]
    const float* __restrict__ obj_ov,
    const int*   __restrict__ obj_idx,
    int*   __restrict__ labels,             // [B,P]
    float* __restrict__ locpart,            // [B*pblk]
    int*   __restrict__ npospart) {
    __shared__ float sred[256];
    __shared__ int   sredi[256];
    const int b   = blockIdx.y;
    const int tid = threadIdx.x;
    const int p   = blockIdx.x * 256 + tid;

    float la = 0.f;
    int   pos = 0;
    if (p < P_) {
        int   idx = obj_idx[b * P_ + p];
        float ov  = obj_ov[b * P_ + p];
        int   lab = (ov < THRESH) ? 0 : tcls[b * O_ + idx];
        labels[b * P_ + p] = lab;
        pos = (lab != 0);
        if (pos) {
            float4 t = ((const float4*)tboxes)[b * O_ + idx];
            float cx = 0.5f * (t.x + t.z), cy = 0.5f * (t.y + t.w);
            float w  = t.z - t.x,          h  = t.w - t.y;
            float4 q = ((const float4*)pboxes)[p];
            float g0 = (cx - q.x) * 10.f / q.z;
            float g1 = (cy - q.y) * 10.f / q.w;
            float g2 = __logf(w / q.z) * 5.f;
            float g3 = __logf(h / q.w) * 5.f;
            float4 pb = ((const float4*)pred_boxes)[b * P_ + p];
            la = fabsf(pb.x - g0) + fabsf(pb.y - g1) +
                 fabsf(pb.z - g2) + fabsf(pb.w - g3);
        }
    }
    sred[tid] = la; sredi[tid] = pos; __syncthreads();
    for (int s = 128; s > 0; s >>= 1) {
        if (tid < s) { sred[tid] += sred[tid + s]; sredi[tid] += sredi[tid + s]; }
        __syncthreads();
    }
    if (tid == 0) {
        int slot = b * gridDim.x + blockIdx.x;
        locpart[slot]  = sred[0];
        npospart[slot] = sredi[0];
    }
}

// ---------------------------------------------------------------- reduce n_pos / loc
__global__ __launch_bounds__(256) void k_reduce1(
    const float* __restrict__ locpart, const int* __restrict__ npospart, int n,
    float* __restrict__ loc_sum, int* __restrict__ npos) {
    __shared__ float sred[256];
    __shared__ int   sredi[256];
    const int tid = threadIdx.x;
    float s = 0.f; int c = 0;
    for (int i = tid; i < n; i += 256) { s += locpart[i]; c += npospart[i]; }
    sred[tid] = s; sredi[tid] = c; __syncthreads();
    for (int st = 128; st > 0; st >>= 1) {
        if (tid < st) { sred[tid] += sred[tid + st]; sredi[tid] += sredi[tid + st]; }
        __syncthreads();
    }
    if (tid == 0) { loc_sum[0] = sred[0]; npos[0] = sredi[0]; }
}

// ---------------------------------------------------------------- cross entropy
// 64 rows x 81 classes per block. Tile is staged global->LDS with CDNA5 async
// b128 loads (ASYNCcnt), exp-row-sums computed with V_WMMA_F32_16X16X4_F32
// (ones B-matrix, exact f32 math).
__global__ __launch_bounds__(256) void k_ce(
    const float* __restrict__ scores,   // [B*P, C]
    const int*   __restrict__ labels,   // [B*P]
    float* __restrict__ ce,             // [B*P]
    float* __restrict__ cepart) {       // [ntiles]
    __shared__ alignas(16) float sdata[TILE * C_];
    __shared__ float smax[TILE];
    __shared__ float ssum[TILE];
    __shared__ float sred[256];
    const int       tid      = threadIdx.x;
    const long long tilebase = (long long)blockIdx.x * (TILE * C_);
    const float*    gsrc     = scores + tilebase;

#if USE_ASYNC_LDS
    for (int i4 = tid; i4 < (TILE * C_) / 4; i4 += 256) {
        unsigned     lds_off = (unsigned)(uintptr_t)&sdata[i4 * 4];
        const float* g       = gsrc + i4 * 4;
        asm volatile("global_load_async_to_lds_b128 %0, %1, off"
                     :: "v"(lds_off), "v"(g) : "memory");
    }
    if (blockIdx.x + 1 < gridDim.x && tid * 32 < TILE * C_)
        __builtin_prefetch(&gsrc[TILE * C_ + tid * 32], 0, 1);
    asm volatile("s_wait_asynccnt 0x0" ::: "memory");
#else
    {
        const float4* g4 = (const float4*)gsrc;
        float4*       s4 = (float4*)sdata;
        for (int i4 = tid; i4 < (TILE * C_) / 4; i4 += 256) s4[i4] = g4[i4];
    }
    if (blockIdx.x + 1 < gridDim.x && tid * 32 < TILE * C_)
        __builtin_prefetch(&gsrc[TILE * C_ + tid * 32], 0, 1);
#endif
    __syncthreads();

    if (tid < TILE) {
        float m = -3.4e38f;
        const float* r = &sdata[tid * C_];
        for (int c = 0; c < C_; ++c) m = fmaxf(m, r[c]);
        smax[tid] = m;
    }
    __syncthreads();

#if HAVE_WMMA_F32X4
    if (tid < 128) {                       // waves 0..3, EXEC all-ones per wave
        const int lane  = tid & 31;
        const int r0    = (tid >> 5) * 16;       // 16 rows per wave
        const int row   = r0 + (lane & 15);      // A f32 16x4: M = lane%16
        const int khalf = (lane >> 4) * 2;       // lanes 0-15 -> K 0,1 ; 16-31 -> K 2,3
        const float m   = smax[row];
        const float* r  = &sdata[row * C_];
        v8f acc = {0.f, 0.f, 0.f, 0.f, 0.f, 0.f, 0.f, 0.f};
        v2f bones; bones.x = 1.f; bones.y = 1.f;  // B = ones (layout independent)
        for (int c = 0; c < C_; c += 4) {         // 21 K-chunks (cols 81..83 zero-pad)
            int c0 = c + khalf, c1 = c0 + 1;
            v2f a;
            a.x = (c0 < C_) ? __expf(r[c0] - m) : 0.f;
            a.y = (c1 < C_) ? __expf(r[c1] - m) : 0.f;
            acc = __builtin_amdgcn_wmma_f32_16x16x4_f32(
                false, a, false, bones, (short)0, acc, false, false);
        }
        // D layout: lane 0 holds M=0..7 (N=0) in acc[0..7]; lane 16 holds M=8..15.
        if (lane == 0)  { for (int j = 0; j < 8; ++j) ssum[r0 + j]     = acc[j]; }
        if (lane == 16) { for (int j = 0; j < 8; ++j) ssum[r0 + 8 + j] = acc[j]; }
    }
#else
    if (tid < TILE) {
        float m = smax[tid], s = 0.f;
        const float* r = &sdata[tid * C_];
        for (int c = 0; c < C_; ++c) s += __expf(r[c] - m);
        ssum[tid] = s;
    }
#endif
    __syncthreads();

    float cepos = 0.f;
    if (tid < TILE) {
        const long long gr = (long long)blockIdx.x * TILE + tid;
        const int lab = labels[gr];
        float cev = smax[tid] + __logf(ssum[tid]) - sdata[tid * C_ + lab];
        ce[gr] = cev;
        if (lab != 0) cepos = cev;
    }
    sred[tid] = cepos; __syncthreads();
    for (int s = 128; s > 0; s >>= 1) {
        if (tid < s) sred[tid] += sred[tid + s];
        __syncthreads();
    }
    if (tid == 0) cepart[blockIdx.x] = sred[0];
}

// ---------------------------------------------------------------- top-k per batch row
// sum of k largest of 8732 values via exact bisection on float bit patterns.
__global__ __launch_bounds__(256) void k_topk(
    const float* __restrict__ ce, const int* __restrict__ npos,
    float* __restrict__ hardpart) {
    __shared__ alignas(16) float row[P_];
    __shared__ float sred[256];
    __shared__ int   sredi[256];
    const int b = blockIdx.x, tid = threadIdx.x;
    {
        const float4* g4 = (const float4*)(ce + b * P_);
        float4*       r4 = (float4*)row;
        for (int i4 = tid; i4 < P_ / 4; i4 += 256) r4[i4] = g4[i4];
    }
    __syncthreads();

    long long k = 3ll * (long long)npos[0];   // n_neg = 3*n_pos (global scalar)
    if (k > P_) k = P_;
    float result = 0.f;

    if (k >= P_) {
        float s = 0.f;
        for (int i = tid; i < P_; i += 256) s += row[i];
        sred[tid] = s; __syncthreads();
        for (int st = 128; st > 0; st >>= 1) {
            if (tid < st) sred[tid] += sred[tid + st];
            __syncthreads();
        }
        result = sred[0];
    } else {
        unsigned lo = 0u, hi = 0x7F800000u;   // ce >= 0: bits are order-monotone
        while (hi - lo > 1u) {
            unsigned mid = (lo + hi) >> 1;
            float t = __uint_as_float(mid);
            int c = 0;
            for (int i = tid; i < P_; i += 256) c += (row[i] >= t);
            sredi[tid] = c; __syncthreads();
            for (int st = 128; st > 0; st >>= 1) {
                if (tid < st) sredi[tid] += sredi[tid + st];
                __syncthreads();
            }
            int cnt = sredi[0]; __syncthreads();
            if ((long long)cnt >= k) lo = mid; else hi = mid;   // uniform update
        }
        float kv = __uint_as_float(lo);       // k-th largest value
        float s = 0.f; int cgt = 0;
        for (int i = tid; i < P_; i += 256) {
            float v = row[i];
            if (v > kv) { s += v; ++cgt; }
        }
        sred[tid] = s; sredi[tid] = cgt; __syncthreads();
        for (int st = 128; st > 0; st >>= 1) {
            if (tid < st) { sred[tid] += sred[tid + st]; sredi[tid] += sredi[tid + st]; }
            __syncthreads();
        }
        result = sred[0] + (float)(k - (long long)sredi[0]) * kv;  // ties at kv
    }
    if (tid == 0) hardpart[b] = result;
}

// ---------------------------------------------------------------- finalize
__global__ __launch_bounds__(256) void k_final(
    const float* __restrict__ cepart, int ncep,
    const float* __restrict__ hardpart,
    const float* __restrict__ loc_sum, const int* __restrict__ npos,
    float* __restrict__ out) {
    __shared__ float sred[256];
    const int tid = threadIdx.x;
    float s = 0.f;
    for (int i = tid; i < ncep; i += 256) s += cepart[i];
    for (int i = tid; i < B_;   i += 256) s += hardpart[i];
    sred[tid] = s; __syncthreads();
    for (int st = 128; st > 0; st >>= 1) {
        if (tid < st) sred[tid] += sred[tid + st];
        __syncthreads();
    }
    if (tid == 0) {
        float np = (float)npos[0];
        out[0] = loc_sum[0] / (np * 4.f) + sred[0] / np;
    }
}

// ---------------------------------------------------------------- launch
extern "C" void kernel_launch(void* const* d_in, const int* in_sizes, int n_in,
                              void* d_out, int out_size, void* d_ws, size_t ws_size,
                              hipStream_t stream) {
    (void)in_sizes; (void)n_in; (void)out_size; (void)ws_size;
    const float* pred_boxes   = (const float*)d_in[0];
    const float* pred_scores  = (const float*)d_in[1];
    const float* true_boxes   = (const float*)d_in[2];
    const int*   true_classes = (const int*)d_in[3];
    const float* pboxes       = (const float*)d_in[4];
    float* out = (float*)d_out;

    const int pblk   = (P_ + 255) / 256;    // 35
    const int nslots = pblk * B_;           // 2240
    const int ntiles = BP / TILE;           // 8732

    char*  ws  = (char*)d_ws;
    size_t off = 0;
    auto carve = [&](size_t bytes) -> void* {
        void* p = ws + off;
        off = (off + bytes + 255) & ~(size_t)255;
        return p;
    };
    float* obj_ov              = (float*)carve((size_t)BP * 4);
    int*   obj_idx             = (int*)  carve((size_t)BP * 4);
    int*   labels              = (int*)  carve((size_t)BP * 4);
    float* ce                  = (float*)carve((size_t)BP * 4);
    unsigned long long* best   = (unsigned long long*)carve((size_t)B_ * O_ * 8);
    float* locpart             = (float*)carve((size_t)nslots * 4);
    int*   npospart            = (int*)  carve((size_t)nslots * 4);
    float* cepart              = (float*)carve((size_t)ntiles * 4);
    float* hardpart            = (float*)carve((size_t)B_ * 4);
    float* loc_sum             = (float*)carve(4);
    int*   npos                = (int*)  carve(4);

    k_init  <<<(B_ * O_ + 255) / 256, 256, 0, stream>>>(best);
    k_match <<<dim3(pblk, B_), 256, 0, stream>>>(true_boxes, pboxes, obj_ov, obj_idx, best);
    k_force <<<B_, 32, 0, stream>>>(best, obj_ov, obj_idx);
    k_labels<<<dim3(pblk, B_), 256, 0, stream>>>(pred_boxes, true_boxes, true_classes,
                                                 pboxes, obj_ov, obj_idx,
                                                 labels, locpart, npospart);
    k_reduce1<<<1, 256, 0, stream>>>(locpart, npospart, nslots, loc_sum, npos);
    k_ce    <<<ntiles, 256, 0, stream>>>(pred_scores, labels, ce, cepart);
    k_topk  <<<B_, 256, 0, stream>>>(ce, npos, hardpart);
    k_final <<<1, 256, 0, stream>>>(cepart, ntiles, hardpart, loc_sum, npos, out);
}